// DeepSetsAPEEncoder_28939489641155
// MI455X (gfx1250) — compile-verified
//
#include <hip/hip_runtime.h>
#include <hip/hip_bf16.h>

// Problem constants (fixed in the reference):
//   B=16 graphs, S_g = 256+16g, N_TOT=6016, IN=16, HID=64, OUT=32,
//   N_PAIRS = 2349056, out = [N_PAIRS, 32] f32.
// start_g      = 8g^2 + 248g
// pair_start_g = 65536g + 4096 g(g-1) + 128 g(g-1)(2g-1)/3

#define IN_DIM 16
#define HID    64
#define OUTD   32
#define N_TOT  6016

typedef __attribute__((ext_vector_type(2))) float v2f;
typedef __attribute__((ext_vector_type(4))) float f32x4;
typedef __attribute__((ext_vector_type(8))) float v8f;

// LDS row stride (floats) for the 16xHID intermediate: 68 = 64+4 makes both the
// C/D-layout stores (bank = 4*row + col) and the A-layout float2 reloads
// (banks {4m+k .. 4m+k+3} across the wave) conflict-free on 64 banks.
#define HS_STRIDE 68

// ---------------------------------------------------------------------------
// Kernel 1: per-node MLP  t = relu(ape@W1 + b1)@W2 + b2  using f32 WMMA.
// One wave (32 threads) per 16-node tile; 6016/16 = 376 blocks.
// A-fragment (16x4 f32): lane l holds A[l&15][2*(l>>4) + {0,1}] in {a.x,a.y}.
// B-fragment (4x16 f32): lane l, comp r holds B[r + 2*(l>>4)][l&15].
// C/D tile (16x16 f32): comp v, lane l holds D[v + 8*(l>>4)][l&15].
// ---------------------------------------------------------------------------
__global__ __launch_bounds__(32) void mlp_wmma_kernel(
    const float* __restrict__ ape, const float* __restrict__ W1,
    const float* __restrict__ b1,  const float* __restrict__ W2,
    const float* __restrict__ b2,  float* __restrict__ t)
{
    __shared__ float Hs[16 * HS_STRIDE];

    const int lane = threadIdx.x;   // 0..31, EXEC all ones (no divergence)
    const int m    = lane & 15;
    const int hi   = lane >> 4;     // 0/1: selects K-pair within a fragment
    const int n0   = blockIdx.x * 16;

    // ---- load A fragments of ape tile: 4 K-groups of 4 --------------------
    v2f a1[4];
#pragma unroll
    for (int kb = 0; kb < 4; ++kb) {
        const float* p = ape + (size_t)(n0 + m) * IN_DIM + 4 * kb + 2 * hi;
        a1[kb] = *(const v2f*)p;    // 8B aligned (even float offset)
    }

    // ---- GEMM1: H = relu(ape@W1 + b1), 4 column tiles of 16 ---------------
#pragma unroll
    for (int nt = 0; nt < 4; ++nt) {
        v8f acc = {};
#pragma unroll
        for (int kb = 0; kb < 4; ++kb) {
            const int krow = 4 * kb + 2 * hi;
            v2f b;
            b.x = W1[(size_t)(krow)     * HID + 16 * nt + m];
            b.y = W1[(size_t)(krow + 1) * HID + 16 * nt + m];
            acc = __builtin_amdgcn_wmma_f32_16x16x4_f32(
                false, a1[kb], false, b, (short)0, acc, false, false);
        }
        const float bias = b1[16 * nt + m];
#pragma unroll
        for (int v = 0; v < 8; ++v) {
            float x = fmaxf(acc[v] + bias, 0.0f);
            const int row = v + 8 * hi;
            Hs[row * HS_STRIDE + 16 * nt + m] = x;
        }
    }
    __syncthreads();   // single wave: compiles to ds-wait (+nop barrier)

    // ---- re-load H as A fragments: 16 K-groups of 4 -----------------------
    v2f a2[16];
#pragma unroll
    for (int kb = 0; kb < 16; ++kb) {
        a2[kb] = *(const v2f*)&Hs[m * HS_STRIDE + 4 * kb + 2 * hi];
    }

    // ---- GEMM2: t = H@W2 + b2, 2 column tiles of 16 -----------------------
#pragma unroll
    for (int nt = 0; nt < 2; ++nt) {
        v8f acc = {};
#pragma unroll
        for (int kb = 0; kb < 16; ++kb) {
            const int krow = 4 * kb + 2 * hi;
            v2f b;
            b.x = W2[(size_t)(krow)     * OUTD + 16 * nt + m];
            b.y = W2[(size_t)(krow + 1) * OUTD + 16 * nt + m];
            acc = __builtin_amdgcn_wmma_f32_16x16x4_f32(
                false, a2[kb], false, b, (short)0, acc, false, false);
        }
        const float bias = b2[16 * nt + m];
#pragma unroll
        for (int v = 0; v < 8; ++v) {
            const int row = v + 8 * hi;
            t[(size_t)(n0 + row) * OUTD + 16 * nt + m] = acc[v] + bias;
        }
    }
}

// ---------------------------------------------------------------------------
// Kernel 2: pair expansion  out[pair(g,i,j), :] = t[start_g+i] + t[start_g+j].
// One block per global node n = start_g + i (6016 blocks x 256 threads).
// Thread keeps a fixed float4 of t[i]; j advances by 32 rows per iteration so
// each iteration stores one contiguous, fully coalesced 4 KB chunk per block.
// Output is write-once -> non-temporal stores; t reads stay cached (770 KB).
// ---------------------------------------------------------------------------
__global__ __launch_bounds__(256) void pair_expand_kernel(
    const float* __restrict__ t, float* __restrict__ out)
{
    const int n = blockIdx.x;           // global node id (uniform -> SALU)

    // recover graph id g and its node start (unrolled, branch-free)
    int g = 0, start = 0;
#pragma unroll
    for (int h = 1; h < 16; ++h) {
        const int s = 8 * h * h + 248 * h;
        if (n >= s) { g = h; start = s; }
    }
    const int S = 256 + 16 * g;
    const int i = n - start;
    const long long ps =
        65536LL * g + 4096LL * g * (g - 1) +
        (128LL * g * (g - 1) * (2 * g - 1)) / 3;
    const long long rowbase = (ps + (long long)i * S) * (long long)OUTD;

    const int tid = threadIdx.x;
    const int c4  = (tid & 7) * 4;      // which float4 of the 32-wide row
    const int j0  = tid >> 3;           // first j row for this thread

    const f32x4 ti = *(const f32x4*)(t + (size_t)n * OUTD + c4);
    const float* tg = t + (size_t)start * OUTD + c4;
    float* ob = out + rowbase + c4;

    for (int j = j0; j < S; j += 32) {
        f32x4 tj = *(const f32x4*)(tg + (size_t)j * OUTD);
        f32x4 r  = ti + tj;
        __builtin_nontemporal_store(r, (f32x4*)(ob + (long long)j * OUTD));
    }
}

// ---------------------------------------------------------------------------
extern "C" void kernel_launch(void* const* d_in, const int* in_sizes, int n_in,
                              void* d_out, int out_size, void* d_ws, size_t ws_size,
                              hipStream_t stream) {
    (void)in_sizes; (void)n_in; (void)out_size; (void)ws_size;
    const float* ape = (const float*)d_in[0];
    const float* W1  = (const float*)d_in[1];
    const float* b1  = (const float*)d_in[2];
    const float* W2  = (const float*)d_in[3];
    const float* b2  = (const float*)d_in[4];
    float* t   = (float*)d_ws;     // [6016, 32] f32 scratch = 770 KB
    float* out = (float*)d_out;    // [2349056, 32] f32

    mlp_wmma_kernel<<<N_TOT / 16, 32, 0, stream>>>(ape, W1, b1, W2, b2, t);
    pair_expand_kernel<<<N_TOT, 256, 0, stream>>>(t, out);
}